// SimpleConfigurablePrimitiveRenderer_7438883357121
// MI455X (gfx1250) — compile-verified
//
#include <hip/hip_runtime.h>
#include <hip/hip_bf16.h>

typedef __attribute__((ext_vector_type(2))) float v2f;
typedef __attribute__((ext_vector_type(8))) float v8f;

#define SZ   256
#define NSEG 64
#define PPB  512            // pixels per block (2 image rows)
#define TPW  (PPB / 16 / 8) // 16-pixel tiles per wave (8 waves/block) = 4

__global__ __launch_bounds__(256)
void SimpleConfigurablePrimitiveRenderer_kernel(const float* __restrict__ shape_params,
                                                const float* __restrict__ sigma_ptr,
                                                float* __restrict__ out)
{
    // Per-segment coefficient table, pre-scaled by c = -log2(e)/sigma so the
    // WMMA outputs are directly s' = c*s and q' = c*q (exp argument units):
    // [0]=c*ba_r [1]=c*ba_c [2]=-c*a.ba [3]=-2c*a_r [4]=-2c*a_c [5]=c*|a|^2
    // [6]=c*baba [7]=(1/max(baba,1e-12))/c   (so u = s'*[7] == s/max(baba,eps))
    __shared__ float seg[NSEG][8];

    const int b   = blockIdx.y;
    const int tid = threadIdx.x;

    const float c  = -1.4426950408889634f / sigma_ptr[0];
    const float rc = 1.0f / c;

    if (tid < NSEG) {
        const float* sp = shape_params + b * (NSEG * 4) + tid * 4;
        float ar = sp[0], ac = sp[1], br = sp[2], bc = sp[3];
        float bar = br - ar, bac = bc - ac;
        float baba   = bar * bar + bac * bac;
        float adotba = ar * bar + ac * bac;
        seg[tid][0] = c * bar;
        seg[tid][1] = c * bac;
        seg[tid][2] = -c * adotba;
        seg[tid][3] = -2.0f * c * ar;
        seg[tid][4] = -2.0f * c * ac;
        seg[tid][5] = c * (ar * ar + ac * ac);
        seg[tid][6] = c * baba;
        seg[tid][7] = rc / fmaxf(baba, 1e-12f);
    }
    __syncthreads();

    const int lane = tid & 31;
    const int half = lane >> 4;   // 0: lanes 0-15, 1: lanes 16-31
    const int col  = lane & 15;   // N index (pixel within tile) / M index for A

    // A matrices: M=segment (lane%16), K striped across lane halves per ISA 16x4 f32 layout.
    // As: s' = f . c*(ba_r, ba_c, 0, -a.ba)   Aq: q' = f . c*(-2a_r, -2a_c, 1, |a|^2)
    v2f As[4], Aq[4];
    float bbt[4][8], ibt[4][8];  // per-lane c*baba / invbaba/c for D rows m = st*16 + half*8 + k
#pragma unroll
    for (int st = 0; st < 4; ++st) {
        int m = (st << 4) | col;
        if (half == 0) {
            As[st][0] = seg[m][0]; As[st][1] = seg[m][1];  // K=0,1
            Aq[st][0] = seg[m][3]; Aq[st][1] = seg[m][4];
        } else {
            As[st][0] = 0.0f;      As[st][1] = seg[m][2];  // K=2,3
            Aq[st][0] = c;         Aq[st][1] = seg[m][5];  // K=2 coeff of q' is c*1
        }
#pragma unroll
        for (int k = 0; k < 8; ++k) {
            int mr = (st << 4) + (half << 3) + k;
            bbt[st][k] = seg[mr][6];
            ibt[st][k] = seg[mr][7];
        }
    }

    const float step = 2.0f / 255.0f;  // linspace(-1,1,256) step
    const int wave = tid >> 5;
    const int blockPixBase = blockIdx.x * PPB;
    const v8f czero = {0.f, 0.f, 0.f, 0.f, 0.f, 0.f, 0.f, 0.f};
    float* outb = out + b * (SZ * SZ);

#pragma unroll
    for (int it = 0; it < TPW; ++it) {
        int tileBase = blockPixBase + (wave * TPW + it) * 16;
        int pix = tileBase + col;
        int h = pix >> 8, w = pix & 255;
        float gr = fmaf((float)h, step, -1.0f);
        float gc = fmaf((float)w, step, -1.0f);
        float gg = gr * gr + gc * gc;
        // B matrix (K=4 x N=16): K striped across lane halves, N = lane%16.
        v2f B;
        B[0] = half ? gg   : gr;  // K=2 : K=0
        B[1] = half ? 1.0f : gc;  // K=3 : K=1

        float prod = 1.0f;  // product over this lane-half's 32 segments, this pixel
#pragma unroll
        for (int st = 0; st < 4; ++st) {
            v8f dsv = __builtin_amdgcn_wmma_f32_16x16x4_f32(
                false, As[st], false, B, (short)0, czero, false, false);
            v8f dqv = __builtin_amdgcn_wmma_f32_16x16x4_f32(
                false, Aq[st], false, B, (short)0, czero, false, false);
#pragma unroll
            for (int k = 0; k < 8; ++k) {
                float s  = dsv[k];                       // c * (pa.ba)
                float q  = dqv[k];                       // c * |pa|^2
                float u  = s * ibt[st][k];               // t before clamp (c cancels)
                float t  = fminf(fmaxf(u, 0.0f), 1.0f);  // fuses to 'clamp' modifier
                // arg = c*d2 = q' - 2*t*s' + t^2 * (c*baba)   (always <= 0)
                float arg = fmaf(t * t, bbt[st][k], fmaf(-2.0f * t, s, q));
                float e   = __builtin_amdgcn_exp2f(arg); // raw v_exp_f32; underflow->0 is exact
                prod      = fmaf(-prod, e, prod);        // prod *= (1 - e)
            }
        }
        // Combine the two lane-halves (segments m and m+8) for the same pixel column.
        float other = __shfl_xor(prod, 16, 32);
        float pall  = prod * other;
        if (half == 0) outb[pix] = 1.0f - pall;
    }
}

extern "C" void kernel_launch(void* const* d_in, const int* in_sizes, int n_in,
                              void* d_out, int out_size, void* d_ws, size_t ws_size,
                              hipStream_t stream) {
    const float* shape = (const float*)d_in[0];   // (bs, 256)
    // d_in[1] colour_params, d_in[2] thickness_params: unused by reference forward
    const float* sigma = (const float*)d_in[3];   // (1,)
    float* out = (float*)d_out;                   // (bs, 1, 256, 256) f32

    int bs = in_sizes[0] / (NSEG * 4);
    dim3 grid(SZ * SZ / PPB, bs);
    SimpleConfigurablePrimitiveRenderer_kernel<<<grid, 256, 0, stream>>>(shape, sigma, out);
}